// AttentionEmbedding_33449205301619
// MI455X (gfx1250) — compile-verified
//
#include <hip/hip_runtime.h>
#include <math.h>

// ---- fixed reference dims (setup_inputs) ----
#define EMBD   512
#define DEVD   448
#define NH     8
#define HD     64
#define MAXD   256
#define LN_EPS 1e-5f
#define LOGK   0.14391156463f   // ln(10000)/64

typedef __attribute__((ext_vector_type(2))) float v2f;
typedef __attribute__((ext_vector_type(8))) float v8f;

static __device__ __forceinline__ v8f wmma_f32(v2f a, v2f b, v8f c) {
    // V_WMMA_F32_16X16X4_F32 : D = A(16x4,f32) * B(4x16,f32) + C(16x16,f32)
    return __builtin_amdgcn_wmma_f32_16x16x4_f32(
        false, a, false, b, (short)0, c, false, false);
}

#define ZERO8 {0.f,0.f,0.f,0.f,0.f,0.f,0.f,0.f}

// ---------------------------------------------------------------- meta
__global__ void init_meta(int* inv, int R, float* zbuf, int Z) {
    int i = blockIdx.x * blockDim.x + threadIdx.x;
    if (i < R) inv[i] = -1;
    if (i < Z) zbuf[i] = 0.f;
}

__global__ void token_meta(const int* __restrict__ si, int B, int N,
                           int* __restrict__ rowmap, float* __restrict__ posf,
                           int* __restrict__ inv) {
    int i = blockIdx.x * blockDim.x + threadIdx.x;
    if (i >= N) return;
    int b = 0;
    for (int bb = 0; bb < B; ++bb)
        if (i >= si[bb] && i < si[bb + 1]) { b = bb; break; }
    int pos0 = i - si[b];
    int r = b * MAXD + pos0;
    rowmap[i] = r;
    posf[i] = (float)(pos0 + 1);     // 1-based position
    inv[r] = i;
}

// ---------------------------------------------------------------- embed GEMM
// emb[i][e] = leaky_relu( sum_c x[i][c]*W1[e][c] + b1[e] ),  x = [states | PE]
// wave computes 16 rows x 64 cols (4 accumulator tiles)
__global__ __launch_bounds__(256)
void embed_gemm(const float* __restrict__ states, const float* __restrict__ W1,
                const float* __restrict__ b1, const float* __restrict__ posf,
                float* __restrict__ emb, float* __restrict__ out) {
    const int CG = EMBD / 64;                           // 8 col groups
    int wave = blockIdx.x * (blockDim.x >> 5) + (threadIdx.x >> 5);
    int rowTile = wave / CG, cg = wave % CG;
    int lane = threadIdx.x & 31, half = lane >> 4, lo = lane & 15;
    int row0 = rowTile * 16, col0 = cg * 64;

    const float* ap = states + (size_t)(row0 + lo) * DEVD + 2 * half;
    const float* bp[4];
#pragma unroll
    for (int t = 0; t < 4; ++t)
        bp[t] = W1 + (size_t)(col0 + 16 * t + lo) * EMBD + 2 * half;

    v8f acc[4] = {ZERO8, ZERO8, ZERO8, ZERO8};

    // ---- states region: k in [0,448), unroll 4 (k-step 16)
    for (int k0 = 0; k0 < DEVD; k0 += 16) {
        v2f A[4], Bv[4][4];
#pragma unroll
        for (int u = 0; u < 4; ++u)
            A[u] = *(const v2f*)(ap + k0 + 4 * u);
#pragma unroll
        for (int t = 0; t < 4; ++t)
#pragma unroll
            for (int u = 0; u < 4; ++u)
                Bv[t][u] = *(const v2f*)(bp[t] + k0 + 4 * u);
#pragma unroll
        for (int u = 0; u < 4; ++u)
#pragma unroll
            for (int t = 0; t < 4; ++t)
                acc[t] = wmma_f32(A[u], Bv[t][u], acc[t]);
    }
    // ---- PE region: k in [448,512), unroll 4
    float pos = posf[row0 + lo];
    for (int k0 = DEVD; k0 < EMBD; k0 += 16) {
        v2f A[4], Bv[4][4];
#pragma unroll
        for (int u = 0; u < 4; ++u) {
            int j = k0 + 4 * u + 2 * half - DEVD;      // even PE col -> sin, +1 -> cos
            float ang = pos * __expf(-(float)j * LOGK);
            A[u][0] = __sinf(ang);
            A[u][1] = __cosf(ang);
        }
#pragma unroll
        for (int t = 0; t < 4; ++t)
#pragma unroll
            for (int u = 0; u < 4; ++u)
                Bv[t][u] = *(const v2f*)(bp[t] + k0 + 4 * u);
#pragma unroll
        for (int u = 0; u < 4; ++u)
#pragma unroll
            for (int t = 0; t < 4; ++t)
                acc[t] = wmma_f32(A[u], Bv[t][u], acc[t]);
    }
    // ---- epilogue
#pragma unroll
    for (int t = 0; t < 4; ++t) {
        int col = col0 + 16 * t + lo;
        float bias = b1[col];
#pragma unroll
        for (int j = 0; j < 8; ++j) {
            int row = row0 + j + 8 * half;
            float v = acc[t][j] + bias;
            v = v > 0.f ? v : 0.01f * v;               // leaky_relu
            emb[(size_t)row * EMBD + col] = v;
            out[(size_t)row * (2 * EMBD) + col] = v;   // output first half
        }
    }
}

// ---------------------------------------------------------------- QKV GEMM
__global__ __launch_bounds__(256)
void qkv_gemm(const float* __restrict__ emb, const int* __restrict__ inv,
              const float* __restrict__ zbuf,
              const float* __restrict__ in_w, const float* __restrict__ in_b,
              float* __restrict__ Q, float* __restrict__ K, float* __restrict__ V) {
    const int CG = (3 * EMBD) / 64;                    // 24 col groups
    int wave = blockIdx.x * (blockDim.x >> 5) + (threadIdx.x >> 5);
    int rowTile = wave / CG, cg = wave % CG;
    int lane = threadIdx.x & 31, half = lane >> 4, lo = lane & 15;
    int row0 = rowTile * 16, col0 = cg * 64;

    int tok = inv[row0 + lo];
    const float* ap = (tok >= 0 ? emb + (size_t)tok * EMBD : zbuf) + 2 * half;
    const float* bp[4];
#pragma unroll
    for (int t = 0; t < 4; ++t)
        bp[t] = in_w + (size_t)(col0 + 16 * t + lo) * EMBD + 2 * half;

    v8f acc[4] = {ZERO8, ZERO8, ZERO8, ZERO8};
    for (int k0 = 0; k0 < EMBD; k0 += 16) {
        v2f A[4], Bv[4][4];
#pragma unroll
        for (int u = 0; u < 4; ++u)
            A[u] = *(const v2f*)(ap + k0 + 4 * u);
#pragma unroll
        for (int t = 0; t < 4; ++t)
#pragma unroll
            for (int u = 0; u < 4; ++u)
                Bv[t][u] = *(const v2f*)(bp[t] + k0 + 4 * u);
#pragma unroll
        for (int u = 0; u < 4; ++u)
#pragma unroll
            for (int t = 0; t < 4; ++t)
                acc[t] = wmma_f32(A[u], Bv[t][u], acc[t]);
    }
#pragma unroll
    for (int t = 0; t < 4; ++t) {
        int n = col0 + 16 * t + lo;
        float bias = in_b[n];
        int seg = n >> 9;                              // 0:Q 1:K 2:V
        int nn = n & (EMBD - 1);
        int h = nn >> 6, d = nn & (HD - 1);
        float* dst = (seg == 0) ? Q : (seg == 1) ? K : V;
#pragma unroll
        for (int j = 0; j < 8; ++j) {
            int r = row0 + j + 8 * half;
            int b = r >> 8, slot = r & (MAXD - 1);
            dst[((((size_t)b * NH + h) * MAXD) + slot) * HD + d] = acc[t][j] + bias;
        }
    }
}

// ---------------------------------------------------------------- attention
// per (batch, head): S = Q Kᵀ / 8, softmax over all 256 keys (no mask), ctx = P V
__global__ __launch_bounds__(256)
void attention(const float* __restrict__ Q, const float* __restrict__ K,
               const float* __restrict__ V, float* __restrict__ ctx) {
    __shared__ float S[32][260];                       // 32-row panel, padded stride
    int bh = blockIdx.x;
    int b = bh >> 3, h = bh & 7;
    const float* Qh = Q + ((size_t)b * NH + h) * MAXD * HD;
    const float* Kh = K + ((size_t)b * NH + h) * MAXD * HD;
    const float* Vh = V + ((size_t)b * NH + h) * MAXD * HD;
    float* ctxh = ctx + (size_t)b * MAXD * EMBD + (size_t)h * HD;

    int wave = threadIdx.x >> 5, lane = threadIdx.x & 31;
    int half = lane >> 4, lo = lane & 15;

    for (int qp = 0; qp < MAXD / 32; ++qp) {           // 8 panels of 32 query rows
        int q0p = qp * 32;
        // ---- scores: each wave computes a 16x64 strip (2 row-grp x 4 col-grp = 8 waves)
        {
            int rt = wave >> 2, cg = wave & 3;
            int qr0 = q0p + rt * 16, kc0 = cg * 64;
            const float* ap = Qh + (size_t)(qr0 + lo) * HD + 2 * half;
            const float* bp[4];
#pragma unroll
            for (int t = 0; t < 4; ++t)
                bp[t] = Kh + (size_t)(kc0 + 16 * t + lo) * HD + 2 * half;
            v8f acc[4] = {ZERO8, ZERO8, ZERO8, ZERO8};
            for (int k0 = 0; k0 < HD; k0 += 16) {      // 4 iterations
                v2f A[4], Bv[4][4];
#pragma unroll
                for (int u = 0; u < 4; ++u)
                    A[u] = *(const v2f*)(ap + k0 + 4 * u);
#pragma unroll
                for (int t = 0; t < 4; ++t)
#pragma unroll
                    for (int u = 0; u < 4; ++u)
                        Bv[t][u] = *(const v2f*)(bp[t] + k0 + 4 * u);
#pragma unroll
                for (int u = 0; u < 4; ++u)
#pragma unroll
                    for (int t = 0; t < 4; ++t)
                        acc[t] = wmma_f32(A[u], Bv[t][u], acc[t]);
            }
#pragma unroll
            for (int t = 0; t < 4; ++t)
#pragma unroll
                for (int j = 0; j < 8; ++j)
                    S[rt * 16 + j + 8 * half][kc0 + 16 * t + lo] = acc[t][j] * 0.125f;
        }
        __syncthreads();
        // ---- softmax: 4 rows per wave, lane-parallel over 256 keys
        for (int rr = 0; rr < 4; ++rr) {
            int row = wave * 4 + rr;
            float vals[8], m = -3.0e38f;
#pragma unroll
            for (int i = 0; i < 8; ++i) { vals[i] = S[row][lane + 32 * i]; m = fmaxf(m, vals[i]); }
            for (int o = 16; o; o >>= 1) m = fmaxf(m, __shfl_xor(m, o, 32));
            float s = 0.f;
#pragma unroll
            for (int i = 0; i < 8; ++i) { vals[i] = __expf(vals[i] - m); s += vals[i]; }
            for (int o = 16; o; o >>= 1) s += __shfl_xor(s, o, 32);
            float is = 1.f / s;
#pragma unroll
            for (int i = 0; i < 8; ++i) S[row][lane + 32 * i] = vals[i] * is;
        }
        __syncthreads();
        // ---- ctx: 16x16 tile per wave (2x4 = 8 tiles), K-dim = 256, unroll 4
        {
            int rt = wave >> 2, ct = wave & 3;
            int qr0 = rt * 16, dc0 = ct * 16;
            const float* srow = &S[qr0 + lo][2 * half];
            const float* vp = Vh + dc0 + lo;
            v8f acc = ZERO8;
            for (int k0 = 0; k0 < MAXD; k0 += 16) {
                v2f A[4], Bv[4];
#pragma unroll
                for (int u = 0; u < 4; ++u)
                    A[u] = *(const v2f*)(srow + k0 + 4 * u);
#pragma unroll
                for (int u = 0; u < 4; ++u)
                    Bv[u] = { vp[(size_t)(k0 + 4 * u + 2 * half) * HD],
                              vp[(size_t)(k0 + 4 * u + 2 * half + 1) * HD] };
#pragma unroll
                for (int u = 0; u < 4; ++u)
                    acc = wmma_f32(A[u], Bv[u], acc);
            }
#pragma unroll
            for (int j = 0; j < 8; ++j) {
                int qg = q0p + qr0 + j + 8 * half;
                ctxh[(size_t)qg * EMBD + dc0 + lo] = acc[j];
            }
        }
        __syncthreads();
    }
}

// ---------------------------------------------------------------- out-proj + residual
__global__ __launch_bounds__(256)
void resid_gemm(const float* __restrict__ ctx, const int* __restrict__ rowmap,
                const float* __restrict__ out_w, const float* __restrict__ out_b,
                const float* __restrict__ emb, float* __restrict__ res) {
    const int CG = EMBD / 64;                          // 8 col groups
    int wave = blockIdx.x * (blockDim.x >> 5) + (threadIdx.x >> 5);
    int rowTile = wave / CG, cg = wave % CG;
    int lane = threadIdx.x & 31, half = lane >> 4, lo = lane & 15;
    int row0 = rowTile * 16, col0 = cg * 64;

    int r = rowmap[row0 + lo];
    const float* ap = ctx + (size_t)r * EMBD + 2 * half;
    const float* bp[4];
#pragma unroll
    for (int t = 0; t < 4; ++t)
        bp[t] = out_w + (size_t)(col0 + 16 * t + lo) * EMBD + 2 * half;

    v8f acc[4] = {ZERO8, ZERO8, ZERO8, ZERO8};
    for (int k0 = 0; k0 < EMBD; k0 += 16) {
        v2f A[4], Bv[4][4];
#pragma unroll
        for (int u = 0; u < 4; ++u)
            A[u] = *(const v2f*)(ap + k0 + 4 * u);
#pragma unroll
        for (int t = 0; t < 4; ++t)
#pragma unroll
            for (int u = 0; u < 4; ++u)
                Bv[t][u] = *(const v2f*)(bp[t] + k0 + 4 * u);
#pragma unroll
        for (int u = 0; u < 4; ++u)
#pragma unroll
            for (int t = 0; t < 4; ++t)
                acc[t] = wmma_f32(A[u], Bv[t][u], acc[t]);
    }
#pragma unroll
    for (int t = 0; t < 4; ++t) {
        int col = col0 + 16 * t + lo;
        float bias = out_b[col];
#pragma unroll
        for (int j = 0; j < 8; ++j) {
            int row = row0 + j + 8 * half;
            res[(size_t)row * EMBD + col] = acc[t][j] + bias + emb[(size_t)row * EMBD + col];
        }
    }
}

// ---------------------------------------------------------------- LayerNorm + gather
__global__ __launch_bounds__(256)
void ln_gather(const float* __restrict__ res, const float* __restrict__ gamma,
               const float* __restrict__ beta, float* __restrict__ out, int N) {
    int row = blockIdx.x * 8 + (threadIdx.x >> 5);
    int lane = threadIdx.x & 31;
    if (row >= N) return;
    const float* r = res + (size_t)row * EMBD;
    float v[16], s = 0.f;
#pragma unroll
    for (int t = 0; t < 16; ++t) { v[t] = r[lane + 32 * t]; s += v[t]; }
    for (int o = 16; o; o >>= 1) s += __shfl_xor(s, o, 32);
    float mu = s * (1.0f / EMBD);
    float s2 = 0.f;
#pragma unroll
    for (int t = 0; t < 16; ++t) { float d = v[t] - mu; s2 += d * d; }
    for (int o = 16; o; o >>= 1) s2 += __shfl_xor(s2, o, 32);
    float rstd = rsqrtf(s2 * (1.0f / EMBD) + LN_EPS);
    float* o2 = out + (size_t)row * (2 * EMBD) + EMBD;
#pragma unroll
    for (int t = 0; t < 16; ++t) {
        int c = lane + 32 * t;
        o2[c] = (v[t] - mu) * rstd * gamma[c] + beta[c];
    }
}

// ---------------------------------------------------------------- launch
extern "C" void kernel_launch(void* const* d_in, const int* in_sizes, int n_in,
                              void* d_out, int out_size, void* d_ws, size_t ws_size,
                              hipStream_t stream) {
    (void)n_in; (void)out_size; (void)ws_size;
    const float* states = (const float*)d_in[0];
    const int*   si     = (const int*)d_in[1];
    const float* W1     = (const float*)d_in[2];
    const float* b1     = (const float*)d_in[3];
    const float* in_w   = (const float*)d_in[4];
    const float* in_b   = (const float*)d_in[5];
    const float* out_w  = (const float*)d_in[6];
    const float* out_b  = (const float*)d_in[7];
    const float* gamma  = (const float*)d_in[8];
    const float* beta   = (const float*)d_in[9];
    float* out = (float*)d_out;

    int N = in_sizes[0] / DEVD;        // 12288
    int B = in_sizes[1] - 1;           // 64
    int R = B * MAXD;                  // 16384 padded dense rows

    char* ws = (char*)d_ws;
    size_t off = 0;
    auto alloc = [&](size_t bytes) -> void* {
        void* p = (void*)(ws + off);
        off += (bytes + 255) & ~(size_t)255;
        return p;
    };
    int*   rowmap = (int*)  alloc((size_t)N * 4);
    float* posf   = (float*)alloc((size_t)N * 4);
    int*   inv    = (int*)  alloc((size_t)R * 4);
    float* zbuf   = (float*)alloc((size_t)1024 * 4);
    float* emb    = (float*)alloc((size_t)N * EMBD * 4);
    float* Qb     = (float*)alloc((size_t)B * NH * MAXD * HD * 4);
    float* Kb     = (float*)alloc((size_t)B * NH * MAXD * HD * 4);
    float* Vb     = (float*)alloc((size_t)B * NH * MAXD * HD * 4);
    float* ctx    = (float*)alloc((size_t)R * EMBD * 4);
    float* res    = (float*)alloc((size_t)N * EMBD * 4);

    init_meta<<<(R + 255) / 256, 256, 0, stream>>>(inv, R, zbuf, 1024);
    token_meta<<<(N + 255) / 256, 256, 0, stream>>>(si, B, N, rowmap, posf, inv);

    int waves1 = (N / 16) * (EMBD / 64);          // 6144 waves, 8/block
    embed_gemm<<<waves1 / 8, 256, 0, stream>>>(states, W1, b1, posf, emb, out);

    int waves2 = (R / 16) * ((3 * EMBD) / 64);    // 24576 waves
    qkv_gemm<<<waves2 / 8, 256, 0, stream>>>(emb, inv, zbuf, in_w, in_b, Qb, Kb, Vb);

    attention<<<B * NH, 256, 0, stream>>>(Qb, Kb, Vb, ctx);

    int waves4 = (N / 16) * (EMBD / 64);
    resid_gemm<<<waves4 / 8, 256, 0, stream>>>(ctx, rowmap, out_w, out_b, emb, res);

    ln_gather<<<N / 8, 256, 0, stream>>>(res, gamma, beta, out, N);
}